// DyDepthwiseConvAtten_6047313952932
// MI455X (gfx1250) — compile-verified
//
#include <hip/hip_runtime.h>

// Problem constants (fixed by reference)
#define BB 1024
#define NN 100
#define CC 256
#define KK 3
#define ROWS (BB * NN)        // 102400 (b,n) rows
#define QSTRIDE 260           // padded LDS row stride in floats (4*l15 bank spread, 16B aligned)
#define WAVES 2
#define TPB (WAVES * 32)
#define ROWS_PER_BLOCK (WAVES * 16)   // 32

typedef __attribute__((ext_vector_type(2))) float v2f;
typedef __attribute__((ext_vector_type(8))) float v8f;

__global__ __launch_bounds__(TPB)
void dydw_conv_ln_kernel(const float* __restrict__ query,
                         const float* __restrict__ value,
                         const float* __restrict__ Ww,     // [K=3, C=256]
                         const float* __restrict__ bw,     // [3]
                         const float* __restrict__ gamma,  // [256]
                         const float* __restrict__ beta,   // [256]
                         float* __restrict__ out)
{
    // B-matrix rows (W_w^T columns) : rows 0..2 = W_w, row 3 = zeros (for N-columns 3..15)
    __shared__ __align__(16) float Wsh[4 * QSTRIDE];
    // 16 query rows per wave, padded stride for conflict-free WMMA A reads
    __shared__ __align__(16) float q_lds[WAVES * 16 * QSTRIDE];
    // dynamic depthwise weights per row: [16 rows][4] per wave
    __shared__ __align__(16) float wt_sh[WAVES * 16 * 4];

    const int tid  = threadIdx.x;
    const int wv   = tid >> 5;       // wave id in block
    const int lane = tid & 31;
    const int l15  = lane & 15;
    const int hi   = lane >> 4;      // lane half (selects K offset in A/B tiles)

    // ---- stage W_w into LDS (rows 0..2) + zero row 3 ----
    for (int idx = tid; idx < KK * CC; idx += TPB)
        Wsh[(idx >> 8) * QSTRIDE + (idx & 255)] = Ww[idx];
    for (int idx = tid; idx < QSTRIDE; idx += TPB)
        Wsh[3 * QSTRIDE + idx] = 0.0f;
    __syncthreads();

    const int rowBase = blockIdx.x * ROWS_PER_BLOCK + wv * 16;

    // ---- phase 1: coalesced stage of 16 query rows into LDS; prefetch value rows ----
    float* qs = &q_lds[wv * 16 * QSTRIDE];
    const float* qbase = query + (size_t)rowBase * CC;
    #pragma unroll
    for (int it = 0; it < 32; ++it) {
        const int f   = it * 32 + lane;     // float4 index within the 16-row tile
        const int row = f >> 6;             // 64 float4 per row
        const int c   = (f & 63) << 2;
        const float4 qv = *(const float4*)(qbase + row * CC + c);
        *(float4*)(qs + row * QSTRIDE + c) = qv;
        // pull matching value rows toward cache while WMMA runs
        __builtin_prefetch(value + (size_t)(rowBase + row) * CC + c, 0, 3);
    }

    // ---- phase 2: dynamic weights via V_WMMA_F32_16X16X4_F32 ----
    // A 16x4 f32 layout: lane l15 -> row M=l15, K = 4j + 2*hi (+0,+1)
    // B 4x16  f32 layout: lane l15 -> col N=l15, same K pair -> contiguous pair of W_w[l15]
    const float* arow  = qs + l15 * QSTRIDE;
    const float* brow  = &Wsh[(l15 < KK ? l15 : 3) * QSTRIDE];
    const int    khalf = hi * 2;

    v8f acc = {};
    #pragma unroll 8
    for (int j = 0; j < CC / 4; ++j) {
        const int kb = j * 4 + khalf;
        const v2f a = *(const v2f*)(arow + kb);   // ds_load_b64, conflict-free
        const v2f b = *(const v2f*)(brow + kb);   // ds_load_b64 (broadcast-ish)
        acc = __builtin_amdgcn_wmma_f32_16x16x4_f32(
                  false, a, false, b, (short)0, acc, false, false);
    }

    // D layout: VGPR r holds M=r (lanes 0-15) / M=r+8 (lanes 16-31); column N = l15.
    float* wt = &wt_sh[wv * 16 * 4];
    if (l15 < KK) {
        const float bwv = bw[l15];
        #pragma unroll
        for (int r = 0; r < 8; ++r)
            wt[(r + 8 * hi) * 4 + l15] = acc[r] + bwv;
    }

    // ---- per-lane gamma/beta (8 channels each) ----
    const float4 g0  = *(const float4*)(gamma + lane * 8);
    const float4 g1  = *(const float4*)(gamma + lane * 8 + 4);
    const float4 be0 = *(const float4*)(beta  + lane * 8);
    const float4 be1 = *(const float4*)(beta  + lane * 8 + 4);

    // ---- phase 3: depthwise 3-tap conv along C + LayerNorm, one row per iteration ----
    for (int m = 0; m < 16; ++m) {
        const float w0 = wt[m * 4 + 0];
        const float w1 = wt[m * 4 + 1];
        const float w2 = wt[m * 4 + 2];

        const size_t gr = (size_t)(rowBase + m) * CC;
        const float4 va = *(const float4*)(value + gr + lane * 8);
        const float4 vb = *(const float4*)(value + gr + lane * 8 + 4);
        float e[8] = {va.x, va.y, va.z, va.w, vb.x, vb.y, vb.z, vb.w};

        // halo exchange across lanes (zero padding at row edges)
        float left  = __shfl_up(vb.w, 1);
        if (lane == 0)  left = 0.0f;
        float right = __shfl_down(va.x, 1);
        if (lane == 31) right = 0.0f;

        float o[8];
        #pragma unroll
        for (int j = 0; j < 8; ++j) {
            const float lft = (j == 0) ? left  : e[j - 1];
            const float rgt = (j == 7) ? right : e[j + 1];
            o[j] = w0 * lft + w1 * e[j] + w2 * rgt;   // out[c] = sum_k w[k]*v[c+k-1]
        }

        // LayerNorm stats: wave32 tree reduction
        float s = 0.0f, s2 = 0.0f;
        #pragma unroll
        for (int j = 0; j < 8; ++j) { s += o[j]; s2 += o[j] * o[j]; }
        #pragma unroll
        for (int off = 16; off > 0; off >>= 1) {
            s  += __shfl_xor(s,  off);
            s2 += __shfl_xor(s2, off);
        }
        const float mu   = s * (1.0f / CC);
        const float var  = s2 * (1.0f / CC) - mu * mu;
        const float rstd = rsqrtf(var + 1e-5f);

        float4 ra, rb;
        ra.x = (o[0] - mu) * rstd * g0.x + be0.x;
        ra.y = (o[1] - mu) * rstd * g0.y + be0.y;
        ra.z = (o[2] - mu) * rstd * g0.z + be0.z;
        ra.w = (o[3] - mu) * rstd * g0.w + be0.w;
        rb.x = (o[4] - mu) * rstd * g1.x + be1.x;
        rb.y = (o[5] - mu) * rstd * g1.y + be1.y;
        rb.z = (o[6] - mu) * rstd * g1.z + be1.z;
        rb.w = (o[7] - mu) * rstd * g1.w + be1.w;

        *(float4*)(out + gr + lane * 8)     = ra;
        *(float4*)(out + gr + lane * 8 + 4) = rb;
    }
}

extern "C" void kernel_launch(void* const* d_in, const int* in_sizes, int n_in,
                              void* d_out, int out_size, void* d_ws, size_t ws_size,
                              hipStream_t stream) {
    const float* query = (const float*)d_in[0];
    const float* value = (const float*)d_in[1];
    const float* Ww    = (const float*)d_in[2];
    const float* bw    = (const float*)d_in[3];
    const float* gamma = (const float*)d_in[4];
    const float* beta  = (const float*)d_in[5];
    float* out = (float*)d_out;

    const int grid = ROWS / ROWS_PER_BLOCK;   // 102400 / 32 = 3200
    dydw_conv_ln_kernel<<<grid, TPB, 0, stream>>>(query, value, Ww, bw, gamma, beta, out);
}